// MSFCN1_CCCA1_71073118814792
// MI455X (gfx1250) — compile-verified
//
#include <hip/hip_runtime.h>
#include <math.h>

// ============================================================================
// MI455X (gfx1250) implementation of MSFCN1+CCCA.
//  - All conv layers run as implicit-GEMM on the CDNA5 WMMA pipe using
//    V_WMMA_F32_16X16X4_F32 (f32 inputs/accum -> matches reference numerics).
//  - The 16384x1501 "grid" and its avg-pool are never materialized: they are
//    pure gathers from x and are fused into conv1's im2col fetch.
//  - One wave (32 threads) computes one 16(out-ch) x 16(position) tile.
// ============================================================================

typedef __attribute__((ext_vector_type(2))) float v2f;
typedef __attribute__((ext_vector_type(8))) float v8f;

// ---------------- static geometry ----------------
constexpr long long XLEN = 12300000LL;       // valid samples; rest of xp is 0
constexpr int GH = 16384, GW = 1501;          // virtual window grid (H, W)

// grid(h,w) = xp[(i/16384)*8192 + (i%16384)],  i = h*1501 + w
__device__ __forceinline__ float grid_at(const float* __restrict__ x, int h, int w) {
  long long i   = (long long)h * GW + w;
  long long blk = i >> 14;          // / 16384
  long long off = i & 16383;        // % 16384
  long long xi  = blk * 8192 + off;
  return (xi < XLEN) ? x[xi] : 0.0f;
}

// AvgPool2d(2,2,pad=1,count_include_pad): out(i,j) covers grid rows 2i-1..2i,
// cols 2j-1..2j (OOB = 0), always divided by 4.
__device__ __forceinline__ float gridp_at(const float* __restrict__ x, int i, int j) {
  const int h0 = 2 * i - 1, w0 = 2 * j - 1;
  float s = 0.0f;
#pragma unroll
  for (int dh = 0; dh < 2; ++dh) {
    const int h = h0 + dh;
    if (h < 0 || h >= GH) continue;
#pragma unroll
    for (int dw = 0; dw < 2; ++dw) {
      const int w = w0 + dw;
      if (w < 0 || w >= GW) continue;
      s += grid_at(x, h, w);
    }
  }
  return 0.25f * s;
}

// im2col fetch. SRC: 0 = plain NCHW buffer, 1 = virtual grid, 2 = pooled grid.
template <int SRC, int CIN, int HIN, int IW, int KH, int KW>
__device__ __forceinline__ float fetch_in(const float* __restrict__ in,
                                          int k, int ih0, int iw0) {
  const int ci = k / (KH * KW);
  const int r  = (k % (KH * KW)) / KW;
  const int s  = k % KW;
  const int ih = ih0 + r, iw = iw0 + s;
  if (ih < 0 || ih >= HIN || iw < 0 || iw >= IW) return 0.0f;
  if constexpr (SRC == 1)      return grid_at(in, ih, iw);
  else if constexpr (SRC == 2) return gridp_at(in, ih, iw);
  else return in[((size_t)ci * HIN + ih) * IW + iw];
}

// ---------------------------------------------------------------------------
// Implicit-GEMM conv via V_WMMA_F32_16X16X4_F32.
//   GEMM: M = COUT (weights, OIHW rows are contiguous K), N = HOUT*WOUT,
//         K = CIN*KH*KW, stepped 4 at a time.
//   A 16x4 layout: lane(0-15)->M=lane, VGPR0/1 = K {k0,k0+1}; lanes 16-31 get
//                  {k0+2,k0+3}.   B 4x16 symmetric with N on lanes.
//   C/D: VGPR g, lane l -> (M = g + 8*(l>=16), N = l&15).
// ---------------------------------------------------------------------------
template <int CIN, int HIN, int IW, int COUT, int HOUT, int OW,
          int KH, int KW, int ST, int PD, bool RELU, bool ACCUM, int SRC>
__global__ __launch_bounds__(32) void conv_wmma_kernel(
    const float* __restrict__ in, const float* __restrict__ wt,
    const float* __restrict__ bias, float* __restrict__ out) {
  constexpr int K    = CIN * KH * KW;
  constexpr int NPOS = HOUT * OW;
  static_assert(K % 4 == 0, "K must be a multiple of 4");
  static_assert(COUT % 16 == 0, "COUT must be a multiple of 16");

  const int lane = threadIdx.x & 31;
  const int half = lane >> 4;    // K-half select (ISA A/B layout)
  const int l16  = lane & 15;
  const int m    = blockIdx.y * 16 + l16;   // A row (out channel) for this lane
  const int n    = blockIdx.x * 16 + l16;   // B col (output position)
  const bool nv  = (n < NPOS);
  const int oh   = nv ? (n / OW) : 0;
  const int ow   = nv ? (n % OW) : 0;
  const int ih0  = oh * ST - PD;
  const int iw0  = ow * ST - PD;

  const float* wrow = wt + (size_t)m * K;
  v8f acc = {};
  for (int k0 = 0; k0 < K; k0 += 4) {
    const int ka = k0 + 2 * half;
    v2f a = *(const v2f*)(wrow + ka);   // 8B-aligned: K%4==0, ka even
    v2f b;
    b.x = nv ? fetch_in<SRC, CIN, HIN, IW, KH, KW>(in, ka,     ih0, iw0) : 0.0f;
    b.y = nv ? fetch_in<SRC, CIN, HIN, IW, KH, KW>(in, ka + 1, ih0, iw0) : 0.0f;
    acc = __builtin_amdgcn_wmma_f32_16x16x4_f32(false, a, false, b,
                                                (short)0, acc, false, false);
  }
  if (nv) {
#pragma unroll
    for (int g = 0; g < 8; ++g) {
      const int mm = blockIdx.y * 16 + g + 8 * half;
      float val = acc[g] + bias[mm];
      if (RELU) val = fmaxf(val, 0.0f);
      const size_t idx = (size_t)mm * NPOS + n;
      if (ACCUM) out[idx] += val; else out[idx] = val;
    }
  }
}

// ---------------------------------------------------------------------------
// Criss-cross attention core (raw-reshape semantics of the reference).
// One thread per output element (c,v,t), joint softmax over C+V+T energies.
// ---------------------------------------------------------------------------
template <int C, int V, int T>
__global__ void cca_attn_kernel(const float* __restrict__ q,
                                const float* __restrict__ k,
                                const float* __restrict__ vv,
                                float* __restrict__ oC, float* __restrict__ oV,
                                float* __restrict__ oT) {
  const int idx = blockIdx.x * blockDim.x + threadIdx.x;
  if (idx >= C * V * T) return;
  const int c = idx / (V * T);
  const int r = idx % (V * T);
  const int v = r / T;
  const int t = r % T;

  const float Qc = q[(size_t)(t * V + v) * C + c];
  const float Qv = q[(size_t)(c * T + t) * V + v];
  const float Qt = q[(size_t)(c * V + v) * T + t];
  const float* Kc = k + (size_t)(t * V + v) * C;
  const float* Kv = k + (size_t)(c * T + t) * V;
  const float* Kt = k + (size_t)(c * V + v) * T;

  float mx = -INFINITY;
  for (int j = 0; j < C; ++j) mx = fmaxf(mx, Qc * Kc[j]);
  for (int j = 0; j < V; ++j) mx = fmaxf(mx, Qv * Kv[j]);
  for (int j = 0; j < T; ++j) mx = fmaxf(mx, Qt * Kt[j]);

  const float* Vc = vv + (size_t)(v * T + t) * C;   // note: scrambled base
  const float* Vv = vv + (size_t)(c * T + t) * V;
  const float* Vt = vv + (size_t)(c * V + v) * T;
  float Z = 0.0f, sC = 0.0f, sV = 0.0f, sT = 0.0f;
  for (int j = 0; j < C; ++j) { const float e = __expf(Qc * Kc[j] - mx); Z += e; sC += e * Vc[j]; }
  for (int j = 0; j < V; ++j) { const float e = __expf(Qv * Kv[j] - mx); Z += e; sV += e * Vv[j]; }
  for (int j = 0; j < T; ++j) { const float e = __expf(Qt * Kt[j] - mx); Z += e; sT += e * Vt[j]; }
  const float inv = 1.0f / Z;
  oC[idx] = sC * inv; oV[idx] = sV * inv; oT[idx] = sT * inv;
}

// a = gamma * sum + x   (elementwise)
__global__ void combine_kernel(const float* __restrict__ sum,
                               const float* __restrict__ xin,
                               const float* __restrict__ gamma,
                               float* __restrict__ out, int n) {
  const int i = blockIdx.x * blockDim.x + threadIdx.x;
  if (i < n) out[i] = gamma[0] * sum[i] + xin[i];
}

// per-channel mean over V*T; one block per channel
__global__ void chan_mean_kernel(const float* __restrict__ a, int VT,
                                 float* __restrict__ out) {
  __shared__ float sm[256];
  const int c = blockIdx.x;
  const float* p = a + (size_t)c * VT;
  float s = 0.0f;
  for (int i = threadIdx.x; i < VT; i += 256) s += p[i];
  sm[threadIdx.x] = s;
  __syncthreads();
  for (int w = 128; w > 0; w >>= 1) {
    if (threadIdx.x < (unsigned)w) sm[threadIdx.x] += sm[threadIdx.x + w];
    __syncthreads();
  }
  if (threadIdx.x == 0) out[c] = sm[0] / (float)VT;
}

// logits[j] = feat . cls_w[j,:] + cls_b[j]
__global__ void final_linear_kernel(const float* __restrict__ feat,
                                    const float* __restrict__ w,
                                    const float* __restrict__ b,
                                    float* __restrict__ out) {
  const int j = threadIdx.x;
  if (j >= 2) return;
  float s = b[j];
  for (int i = 0; i < 256; ++i) s += feat[i] * w[j * 256 + i];
  out[j] = s;
}

// ---------------- workspace layout (floats) ----------------
constexpr size_t SZ_A   = 43870784;           // 64*2731*251 (largest)
constexpr size_t O_A    = 0;
constexpr size_t O_B    = O_A + SZ_A;
constexpr size_t SZ_B   = 5507712;            // 128*683*63
constexpr size_t O_C0   = O_B + SZ_B;         // c0: 128*342*32 = 1400832
constexpr size_t O_C1   = O_C0 + 1400832;     // c1: 128*171*16 = 350208
constexpr size_t O_MEAN = O_C1 + 350208;      // 256 floats
// total: 51,129,792 floats ~ 204.5 MB

extern "C" void kernel_launch(void* const* d_in, const int* in_sizes, int n_in,
                              void* d_out, int out_size, void* d_ws, size_t ws_size,
                              hipStream_t stream) {
  (void)in_sizes; (void)n_in; (void)out_size; (void)ws_size;
  const float* x    = (const float*)d_in[0];
  const float* f0w1 = (const float*)d_in[1];  const float* f0b1 = (const float*)d_in[2];
  const float* f0w2 = (const float*)d_in[3];  const float* f0b2 = (const float*)d_in[4];
  const float* f0w3 = (const float*)d_in[5];  const float* f0b3 = (const float*)d_in[6];
  const float* f0w4 = (const float*)d_in[7];  const float* f0b4 = (const float*)d_in[8];
  const float* f1w1 = (const float*)d_in[9];  const float* f1b1 = (const float*)d_in[10];
  const float* f1w2 = (const float*)d_in[11]; const float* f1b2 = (const float*)d_in[12];
  const float* f1w3 = (const float*)d_in[13]; const float* f1b3 = (const float*)d_in[14];
  const float* f1w4 = (const float*)d_in[15]; const float* f1b4 = (const float*)d_in[16];
  const float* a_wq = (const float*)d_in[17]; const float* a_bq = (const float*)d_in[18];
  const float* a_wk = (const float*)d_in[19]; const float* a_bk = (const float*)d_in[20];
  const float* a_wv = (const float*)d_in[21]; const float* a_bv = (const float*)d_in[22];
  const float* a_wrc = (const float*)d_in[23]; const float* a_brc = (const float*)d_in[24];
  const float* a_wrv = (const float*)d_in[25]; const float* a_brv = (const float*)d_in[26];
  const float* a_wrt = (const float*)d_in[27]; const float* a_brt = (const float*)d_in[28];
  const float* a_gamma = (const float*)d_in[29];
  const float* cls_w = (const float*)d_in[30]; const float* cls_b = (const float*)d_in[31];
  float* ws  = (float*)d_ws;
  float* out = (float*)d_out;

#define LAUNCH_CONV(CIN, HIN, IW, COUT, HOUT, OW, KH, KW, ST, PD, RELU, ACCUM, SRC, IN, W, B, OUT)  \
  do {                                                                                              \
    constexpr int NP_ = (HOUT) * (OW);                                                              \
    dim3 g_((NP_ + 15) / 16, (COUT) / 16);                                                          \
    conv_wmma_kernel<CIN, HIN, IW, COUT, HOUT, OW, KH, KW, ST, PD, RELU, ACCUM, SRC>                \
        <<<g_, dim3(32), 0, stream>>>(IN, W, B, OUT);                                               \
  } while (0)

  // ---- head 0 (grid gather fused into conv1) ----
  LAUNCH_CONV(1, 16384, 1501,  64, 2731, 251, 12, 12, 6, 6, true, false, 1, x,        f0w1, f0b1, ws + O_A);
  LAUNCH_CONV(64, 2731, 251, 128,  683,  63,  8,  8, 4, 4, true, false, 0, ws + O_A,  f0w2, f0b2, ws + O_B);
  LAUNCH_CONV(128, 683,  63, 256,  342,  32,  5,  5, 2, 2, true, false, 0, ws + O_B,  f0w3, f0b3, ws + O_A);
  LAUNCH_CONV(256, 342,  32, 128,  342,  32,  3,  3, 1, 1, true, false, 0, ws + O_A,  f0w4, f0b4, ws + O_C0);

  // ---- head 1 (avgpool fused into conv1) ----
  LAUNCH_CONV(1, 8193,  751,  64, 1366, 126, 12, 12, 6, 6, true, false, 2, x,         f1w1, f1b1, ws + O_A);
  LAUNCH_CONV(64, 1366, 126, 128,  342,  32,  8,  8, 4, 4, true, false, 0, ws + O_A,  f1w2, f1b2, ws + O_B);
  LAUNCH_CONV(128, 342,  32, 256,  171,  16,  5,  5, 2, 2, true, false, 0, ws + O_B,  f1w3, f1b3, ws + O_A);
  LAUNCH_CONV(256, 171,  16, 128,  171,  16,  3,  3, 1, 1, true, false, 0, ws + O_A,  f1w4, f1b4, ws + O_C1);

  // ---- CCA on head 0 output (C=16, V=342, T=32), scratch in region A ----
  {
    constexpr size_t E = 175104;  // 16*342*32
    float* q0 = ws + O_A;            float* k0 = q0 + E;   float* v0 = k0 + E;
    float* oC = v0 + E;              float* oV = oC + E;   float* oT = oV + E;
    float* sum0 = oT + E;            float* a0 = sum0 + 1400832;
    LAUNCH_CONV(128, 342, 32, 16, 342, 32, 3, 3, 1, 1, false, false, 0, ws + O_C0, a_wq, a_bq, q0);
    LAUNCH_CONV(128, 342, 32, 16, 342, 32, 3, 3, 1, 1, false, false, 0, ws + O_C0, a_wk, a_bk, k0);
    LAUNCH_CONV(128, 342, 32, 16, 342, 32, 3, 3, 1, 1, false, false, 0, ws + O_C0, a_wv, a_bv, v0);
    cca_attn_kernel<16, 342, 32><<<(E + 255) / 256, 256, 0, stream>>>(q0, k0, v0, oC, oV, oT);
    LAUNCH_CONV(16, 342, 32, 128, 342, 32, 3, 3, 1, 1, false, false, 0, oC, a_wrc, a_brc, sum0);
    LAUNCH_CONV(16, 342, 32, 128, 342, 32, 3, 3, 1, 1, false, true,  0, oV, a_wrv, a_brv, sum0);
    LAUNCH_CONV(16, 342, 32, 128, 342, 32, 3, 3, 1, 1, false, true,  0, oT, a_wrt, a_brt, sum0);
    combine_kernel<<<(1400832 + 255) / 256, 256, 0, stream>>>(sum0, ws + O_C0, a_gamma, a0, 1400832);
    chan_mean_kernel<<<128, 256, 0, stream>>>(a0, 342 * 32, ws + O_MEAN);
  }

  // ---- CCA on head 1 output (C=16, V=171, T=16), scratch in region B ----
  {
    constexpr size_t E = 43776;   // 16*171*16
    float* q1 = ws + O_B;            float* k1 = q1 + E;   float* v1 = k1 + E;
    float* oC = v1 + E;              float* oV = oC + E;   float* oT = oV + E;
    float* sum1 = oT + E;            float* a1 = sum1 + 350208;
    LAUNCH_CONV(128, 171, 16, 16, 171, 16, 3, 3, 1, 1, false, false, 0, ws + O_C1, a_wq, a_bq, q1);
    LAUNCH_CONV(128, 171, 16, 16, 171, 16, 3, 3, 1, 1, false, false, 0, ws + O_C1, a_wk, a_bk, k1);
    LAUNCH_CONV(128, 171, 16, 16, 171, 16, 3, 3, 1, 1, false, false, 0, ws + O_C1, a_wv, a_bv, v1);
    cca_attn_kernel<16, 171, 16><<<(E + 255) / 256, 256, 0, stream>>>(q1, k1, v1, oC, oV, oT);
    LAUNCH_CONV(16, 171, 16, 128, 171, 16, 3, 3, 1, 1, false, false, 0, oC, a_wrc, a_brc, sum1);
    LAUNCH_CONV(16, 171, 16, 128, 171, 16, 3, 3, 1, 1, false, true,  0, oV, a_wrv, a_brv, sum1);
    LAUNCH_CONV(16, 171, 16, 128, 171, 16, 3, 3, 1, 1, false, true,  0, oT, a_wrt, a_brt, sum1);
    combine_kernel<<<(350208 + 255) / 256, 256, 0, stream>>>(sum1, ws + O_C1, a_gamma, a1, 350208);
    chan_mean_kernel<<<128, 256, 0, stream>>>(a1, 171 * 16, ws + O_MEAN + 128);
  }

  // ---- classifier ----
  final_linear_kernel<<<1, 32, 0, stream>>>(ws + O_MEAN, cls_w, cls_b, out);
#undef LAUNCH_CONV
}